// NTMMemory_69337952026678
// MI455X (gfx1250) — compile-verified
//
#include <hip/hip_runtime.h>
#include <math.h>
#include <stdint.h>

#define B_ 64
#define N_ 8192
#define M_ 128
#define EPSF 1e-16f
#define COS_EPSF 1e-8f
#define ROWSTRIDE 132   // padded LDS row stride (floats): 528B = 33 x 16B, bank-conflict free

typedef __attribute__((ext_vector_type(2))) float v2f;
typedef __attribute__((ext_vector_type(8))) float v8f;

// ---------------------------------------------------------------------------
// Kernel 0: knorm[b] = max(||k[b]+eps||, 1e-8)
// ---------------------------------------------------------------------------
__global__ void ntm_knorm_kernel(const float* __restrict__ k, float* __restrict__ knorm) {
    __shared__ float red[128];
    int b = blockIdx.x, t = threadIdx.x;
    float kv = k[b * M_ + t] + EPSF;
    red[t] = kv * kv;
    __syncthreads();
    for (int s = 64; s > 0; s >>= 1) {
        if (t < s) red[t] += red[t + s];
        __syncthreads();
    }
    if (t == 0) knorm[b] = fmaxf(sqrtf(red[0]), COS_EPSF);
}

// ---------------------------------------------------------------------------
// Kernel 1: content addressing via V_WMMA_F32_16X16X4_F32.
// Each wave32 stages its 16x128 f32 memory tile into LDS with
// GLOBAL_LOAD_ASYNC_TO_LDS_B128 (16 coalesced 512B transactions, ASYNCcnt),
// waits with s_wait_asynccnt, then accumulates 32 WMMA steps (K=4) against
// the k-vector broadcast across all 16 B-columns. Row sum-of-squares for the
// cosine denominator rides along on the VALU from the LDS reads.
// score[b,n] = beta * dot / (max(||row||,1e-8) * knorm[b])
// ---------------------------------------------------------------------------
__global__ void ntm_score_kernel(const float* __restrict__ memory,
                                 const float* __restrict__ k,
                                 const float* __restrict__ beta,
                                 const float* __restrict__ knorm,
                                 float* __restrict__ score) {
    __shared__ float kk[M_];
    __shared__ __align__(16) float tile[4 * 16 * ROWSTRIDE];   // 4 waves x 16 rows
    int b = blockIdx.y;
    int t = threadIdx.x;
    int lane = t & 31;
    int wv = t >> 5;           // wave within block (4 waves, 16 rows each)
    if (t < M_) kk[t] = k[b * M_ + t] + EPSF;

    int n0 = blockIdx.x * 64 + wv * 16;
    const float* gbase = memory + ((size_t)b * N_ + n0) * (size_t)M_;
    float* wtile = tile + wv * 16 * ROWSTRIDE;
    uint32_t lds_base = (uint32_t)(uintptr_t)(void*)wtile;

    // Stage 16 rows x 512B into LDS: iteration i copies row i, 32 lanes x 16B.
#pragma unroll
    for (int i = 0; i < 16; ++i) {
        const void* src = (const void*)(gbase + (size_t)i * M_ + lane * 4);
        uint32_t dst = lds_base + (uint32_t)(i * ROWSTRIDE + lane * 4) * 4u;
        asm volatile("global_load_async_to_lds_b128 %0, %1, off"
                     :: "v"(dst), "v"(src) : "memory");
    }
#if __has_builtin(__builtin_amdgcn_s_wait_asynccnt)
    __builtin_amdgcn_s_wait_asynccnt(0);
#else
    asm volatile("s_wait_asynccnt 0x0" ::: "memory");
#endif
    __syncthreads();   // kk[] visibility across waves

    int row = lane & 15;
    int half = lane >> 4;      // A/B operand K = vgpr + 2*half
    const float* lrow = wtile + row * ROWSTRIDE;

    v8f acc = {0.f, 0.f, 0.f, 0.f, 0.f, 0.f, 0.f, 0.f};
    float sumsq = 0.f;
#pragma unroll
    for (int ks = 0; ks < 32; ++ks) {
        int c = ks * 4 + half * 2;
        float a0 = lrow[c] + EPSF;
        float a1 = lrow[c + 1] + EPSF;
        v2f A = {a0, a1};
        v2f Bv = {kk[c], kk[c + 1]};
        sumsq = fmaf(a0, a0, sumsq);
        sumsq = fmaf(a1, a1, sumsq);
        acc = __builtin_amdgcn_wmma_f32_16x16x4_f32(
            /*neg_a=*/false, A, /*neg_b=*/false, Bv,
            /*c_mod=*/(short)0, acc, /*reuse_a=*/false, /*reuse_b=*/false);
    }
    // combine the two K-halves of each row's sum of squares (wave32 halves)
    sumsq += __shfl_xor(sumsq, 16);
    float nrm = fmaxf(sqrtf(sumsq), COS_EPSF);   // lane l holds norm of row l&15
    float scale = beta[b] / knorm[b];

    // D layout: acc[r] at lane 0  -> D[r][0]   (rows 0..7)
    //           acc[r] at lane 16 -> D[r+8][0] (rows 8..15)
    float out8[8];
#pragma unroll
    for (int r = 0; r < 8; ++r) {
        float nr = __shfl(nrm, r + half * 8);    // norm of row (r + 8*half)
        out8[r] = scale * acc[r] / nr;
    }
    if ((lane & 15) == 0) {
        float* sp = score + (size_t)b * N_ + n0 + half * 8;
#pragma unroll
        for (int r = 0; r < 8; ++r) sp[r] = out8[r];
    }
}

// ---------------------------------------------------------------------------
// Kernel 2: per-b pipeline over N in LDS:
// softmax(score) -> interpolate with w_prev -> circular 3-tap shift ->
// pow(gamma) -> normalize. One 256-thread block per b, fixed reduction order.
// ---------------------------------------------------------------------------
__global__ void ntm_weight_kernel(const float* __restrict__ score,
                                  const float* __restrict__ g,
                                  const float* __restrict__ s,
                                  const float* __restrict__ gamma,
                                  const float* __restrict__ w_prev,
                                  float* __restrict__ w_out) {
    const int TPB = 256, PER = N_ / 256;   // 32 elements per thread
    __shared__ float buf[N_];              // 32 KB
    __shared__ float red[256];
    int b = blockIdx.x, t = threadIdx.x;

    // max
    float lmax = -3.402823e38f;
    for (int j = 0; j < PER; ++j) {
        int i = t + j * TPB;
        float v = score[(size_t)b * N_ + i];
        buf[i] = v;
        lmax = fmaxf(lmax, v);
    }
    red[t] = lmax;
    __syncthreads();
    for (int st = 128; st > 0; st >>= 1) {
        if (t < st) red[t] = fmaxf(red[t], red[t + st]);
        __syncthreads();
    }
    float mx = red[0];
    __syncthreads();

    // exp + sum
    float lsum = 0.f;
    for (int j = 0; j < PER; ++j) {
        int i = t + j * TPB;
        float p = expf(buf[i] - mx);
        buf[i] = p;
        lsum += p;
    }
    red[t] = lsum;
    __syncthreads();
    for (int st = 128; st > 0; st >>= 1) {
        if (t < st) red[t] += red[t + st];
        __syncthreads();
    }
    float sm = red[0];
    __syncthreads();

    // interpolate -> buf holds w_g
    float gb = g[b];
    float s0 = s[b * 3 + 0], s1 = s[b * 3 + 1], s2 = s[b * 3 + 2];
    float gm = gamma[b];
    for (int j = 0; j < PER; ++j) {
        int i = t + j * TPB;
        float wr = buf[i] / sm;
        buf[i] = gb * wr + (1.f - gb) * w_prev[(size_t)b * N_ + i];
    }
    __syncthreads();

    // circular shift + sharpen
    float wp[PER];
    float lps = 0.f;
#pragma unroll
    for (int j = 0; j < PER; ++j) {
        int i = t + j * TPB;
        int il = (i == 0) ? (N_ - 1) : (i - 1);
        int ir = (i == N_ - 1) ? 0 : (i + 1);
        float wsv = s0 * buf[il] + s1 * buf[i] + s2 * buf[ir];
        float p = powf(wsv, gm);
        wp[j] = p;
        lps += p;
    }
    red[t] = lps;
    __syncthreads();
    for (int st = 128; st > 0; st >>= 1) {
        if (t < st) red[t] += red[t + st];
        __syncthreads();
    }
    float inv = 1.f / (red[0] + EPSF);
#pragma unroll
    for (int j = 0; j < PER; ++j) {
        int i = t + j * TPB;
        w_out[(size_t)b * N_ + i] = wp[j] * inv;
    }
}

// ---------------------------------------------------------------------------
// Kernel 3: streaming erase/add write + per-tile partial read sums.
// Grid (tile=N/64, b); block 256: 32 col-groups (float4) x 8 row-threads.
// ---------------------------------------------------------------------------
__global__ void ntm_write_kernel(const float* __restrict__ memory,
                                 const float* __restrict__ w,
                                 const float* __restrict__ e,
                                 const float* __restrict__ a,
                                 float* __restrict__ mem_out,
                                 float* __restrict__ partial) {
    __shared__ float4 sacc[256];
    int b = blockIdx.y, tile = blockIdx.x, t = threadIdx.x;
    int cg = t & 31;      // column group of 4 floats
    int rs = t >> 5;      // 0..7 row sub-slot
    int c4 = cg * 4;

    float4 e4 = ((const float4*)(e + b * M_))[cg];
    float4 a4 = ((const float4*)(a + b * M_))[cg];
    float4 racc = {0.f, 0.f, 0.f, 0.f};

#pragma unroll
    for (int j = 0; j < 8; ++j) {
        int row = tile * 64 + rs + j * 8;
        float wn = w[(size_t)b * N_ + row];
        size_t off = ((size_t)b * N_ + row) * (size_t)M_ + c4;
        float4 mv = *(const float4*)(memory + off);
        float4 nm;
        nm.x = mv.x * (1.f - wn * e4.x) + wn * a4.x;
        nm.y = mv.y * (1.f - wn * e4.y) + wn * a4.y;
        nm.z = mv.z * (1.f - wn * e4.z) + wn * a4.z;
        nm.w = mv.w * (1.f - wn * e4.w) + wn * a4.w;
        *(float4*)(mem_out + off) = nm;
        racc.x += wn * mv.x;
        racc.y += wn * mv.y;
        racc.z += wn * mv.z;
        racc.w += wn * mv.w;
    }
    sacc[t] = racc;
    __syncthreads();
    if (t < 32) {
        float4 sum = sacc[t];
        for (int r = 1; r < 8; ++r) {
            float4 o = sacc[t + r * 32];
            sum.x += o.x; sum.y += o.y; sum.z += o.z; sum.w += o.w;
        }
        *(float4*)(partial + ((size_t)b * 128 + tile) * (size_t)M_ + c4) = sum;
    }
}

// ---------------------------------------------------------------------------
// Kernel 4: read[b,m] = sum over 128 tile partials (fixed order).
// ---------------------------------------------------------------------------
__global__ void ntm_read_reduce_kernel(const float* __restrict__ partial,
                                       float* __restrict__ read_out) {
    int b = blockIdx.x, m = threadIdx.x;   // 128 threads
    float sum = 0.f;
    for (int tile = 0; tile < 128; ++tile)
        sum += partial[((size_t)b * 128 + tile) * (size_t)M_ + m];
    read_out[b * M_ + m] = sum;
}

// ---------------------------------------------------------------------------
extern "C" void kernel_launch(void* const* d_in, const int* in_sizes, int n_in,
                              void* d_out, int out_size, void* d_ws, size_t ws_size,
                              hipStream_t stream) {
    const float* memory = (const float*)d_in[0];
    const float* k      = (const float*)d_in[1];
    const float* beta   = (const float*)d_in[2];
    const float* g      = (const float*)d_in[3];
    const float* s      = (const float*)d_in[4];
    const float* gamma  = (const float*)d_in[5];
    const float* w_prev = (const float*)d_in[6];
    const float* e      = (const float*)d_in[7];
    const float* a      = (const float*)d_in[8];

    float* out      = (float*)d_out;
    float* read_out = out;                                   // [B,M]
    float* w_out    = out + B_ * M_;                         // [B,N]
    float* mem_out  = out + B_ * M_ + (size_t)B_ * N_;       // [B,N,M]

    float* wsf     = (float*)d_ws;
    float* score   = wsf;                                    // B*N floats
    float* knorm   = wsf + (size_t)B_ * N_;                  // B floats
    float* partial = wsf + (size_t)B_ * N_ + 64;             // B*128*M floats

    ntm_knorm_kernel<<<B_, 128, 0, stream>>>(k, knorm);

    dim3 g1(N_ / 64, B_);
    ntm_score_kernel<<<g1, 128, 0, stream>>>(memory, k, beta, knorm, score);

    ntm_weight_kernel<<<B_, 256, 0, stream>>>(score, g, s, gamma, w_prev, w_out);

    dim3 g3(N_ / 64, B_);
    ntm_write_kernel<<<g3, 256, 0, stream>>>(memory, w_out, e, a, mem_out, partial);

    ntm_read_reduce_kernel<<<B_, 128, 0, stream>>>(partial, read_out);
}